// CIF_3272765080394
// MI455X (gfx1250) — compile-verified
//
#include <hip/hip_runtime.h>
#include <math.h>

typedef float v2f __attribute__((ext_vector_type(2)));
typedef float v8f __attribute__((ext_vector_type(8)));

#define B_  32
#define T_  2048
#define H_  256
#define L_  (T_ + 1)
#define C2_ (2 * H_)   /* 512 */

/* workspace layout (float units) */
#define WS_V1   0
#define WS_VS   256
#define WS_CC   512
#define WS_P    1024
#define WS_Q    (1024 + B_ * T_)
#define WS_FLG  (1024 + 2 * B_ * T_)   /* int flags */

/* output layout (float units), concatenated in reference return order */
#define O_CIF   0
#define O_ORI   (B_ * L_ * H_)
#define O_FF    (O_ORI + B_ * T_)
#define O_NP    (O_FF + B_ * T_)
#define O_SUMA  (O_NP + B_ * L_)

/* ------------------------------------------------------------------ */
/* Kernel 1: fuse the two linear layers.                              */
/* col[i]  = sum_o W2[o] * W1[o*512 + i]                              */
/* v1[c]   = col[256+c]          (pairs with -hist part)              */
/* vsum[c] = col[c]+col[256+c]   (pairs with x part)                  */
/* cc      = b2 + sum_o W2[o]*b1[o]                                   */
/* ------------------------------------------------------------------ */
__global__ void cif_prep(const float* __restrict__ W1, const float* __restrict__ b1,
                         const float* __restrict__ W2, const float* __restrict__ b2,
                         float* __restrict__ ws) {
  __shared__ float col[C2_];
  __shared__ float w2s[C2_];
  __shared__ float red[C2_];
  int tid = threadIdx.x;          /* blockDim = 512 */
  w2s[tid] = W2[tid];
  __syncthreads();
  float s = 0.f;
  for (int o = 0; o < C2_; ++o) s = fmaf(w2s[o], W1[(size_t)o * C2_ + tid], s);
  col[tid] = s;
  red[tid] = w2s[tid] * b1[tid];
  __syncthreads();
  for (int off = 256; off; off >>= 1) {
    if (tid < off) red[tid] += red[tid + off];
    __syncthreads();
  }
  if (tid < H_) {
    ws[WS_V1 + tid] = col[H_ + tid];
    ws[WS_VS + tid] = col[tid] + col[H_ + tid];
  }
  if (tid == 0) ws[WS_CC] = red[0] + b2[0];
}

/* ------------------------------------------------------------------ */
/* Kernel 2: per-row dual dot products via V_WMMA_F32_16X16X4_F32.    */
/* One wave per 16-row tile. A = x rows (16x4 f32 tile); B is staged  */
/* in LDS: column 0 -> v1, column 1 -> vsum, columns 2..15 -> zeros,  */
/* selected branchlessly per lane via a base-offset cndmask. Inner    */
/* loop: global_load_b64 (A) + ds_load_b64 (B) + wmma, no EXEC ops.   */
/* D[:,0]=p, D[:,1]=q. Also accumulates sum(|x|) per row for flags.   */
/* ------------------------------------------------------------------ */
__global__ void cif_dots(const float* __restrict__ x, float* __restrict__ ws) {
  float* P = ws + WS_P;
  float* Q = ws + WS_Q;
  int*   FLG = (int*)(ws + WS_FLG);

  __shared__ float sB[3 * H_];   /* [0,256)=v1, [256,512)=vsum, [512,768)=0 */
  __shared__ float sabs[32];

  const int lane = threadIdx.x & 31;       /* blockDim = 32 (one wave) */
  const int tile = blockIdx.x;             /* 0 .. B*T/16-1 */
  const int b  = tile >> 7;                /* T/16 = 128 tiles per batch */
  const int t0 = (tile & 127) << 4;
  const int m  = lane & 15;                /* row in tile / column index */
  const int kh = lane >> 4;                /* 0: K=0,1  1: K=2,3 */

  /* stage B source vectors into LDS (L2-resident, tiny) */
  for (int i = lane; i < H_; i += 32) {
    sB[i]          = ws[WS_V1 + i];
    sB[H_ + i]     = ws[WS_VS + i];
    sB[2 * H_ + i] = 0.f;
  }
  __syncthreads();

  const float* row = x + ((size_t)(b * T_ + t0 + m)) * H_;
  /* branchless per-lane B base: col0 -> v1, col1 -> vsum, else zeros */
  const float* bp = &sB[(m == 0) ? 0 : ((m == 1) ? H_ : 2 * H_)];

#if __has_builtin(__builtin_amdgcn_wmma_f32_16x16x4_f32)
  v8f acc = (v8f)0.f;
  float absacc = 0.f;
#pragma unroll 4
  for (int kc = 0; kc < H_ / 4; ++kc) {
    const int k = kc * 4 + kh * 2;
    const v2f a  = *(const v2f*)(row + k);
    const v2f bb = *(const v2f*)(bp + k);
    absacc += fabsf(a.x) + fabsf(a.y);
    /* (neg_a, A, neg_b, B, c_mod, C, reuse_a, reuse_b) */
    acc = __builtin_amdgcn_wmma_f32_16x16x4_f32(false, a, false, bb,
                                                (short)0, acc, false, false);
  }
  sabs[lane] = absacc;
  __syncthreads();
  if (lane < 16) FLG[b * T_ + t0 + m] = ((sabs[m] + sabs[m + 16]) != 0.f) ? 1 : 0;
  /* C/D layout: VGPR r holds M=r (lanes 0-15, N=lane) and M=r+8 (lanes 16-31) */
  if (lane == 0)  { for (int r = 0; r < 8; ++r) P[b * T_ + t0 + r]     = acc[r]; }
  if (lane == 16) { for (int r = 0; r < 8; ++r) P[b * T_ + t0 + 8 + r] = acc[r]; }
  if (lane == 1)  { for (int r = 0; r < 8; ++r) Q[b * T_ + t0 + r]     = acc[r]; }
  if (lane == 17) { for (int r = 0; r < 8; ++r) Q[b * T_ + t0 + 8 + r] = acc[r]; }
#else
  /* scalar fallback: lanes 0..15 each handle one row */
  if (lane < 16) {
    float p = 0.f, q = 0.f, aa = 0.f;
    for (int k = 0; k < H_; ++k) {
      float xv = row[k];
      p = fmaf(xv, sB[k], p);
      q = fmaf(xv, sB[H_ + k], q);
      aa += fabsf(xv);
    }
    P[b * T_ + t0 + m] = p;
    Q[b * T_ + t0 + m] = q;
    FLG[b * T_ + t0 + m] = (aa != 0.f) ? 1 : 0;
  }
  (void)sabs; (void)kh; (void)bp;
#endif
}

/* ------------------------------------------------------------------ */
/* Kernel 3: one block per batch. Weights, scaling, fire scan,        */
/* integrate, scatter + normalize, all outputs.                       */
/* ------------------------------------------------------------------ */
__global__ __launch_bounds__(256)
void cif_scan(const float* __restrict__ x, const float* __restrict__ mask,
              const int* __restrict__ is_training,
              const float* __restrict__ ws, float* __restrict__ out) {
  __shared__ float s_w[T_];
  __shared__ float s_p[T_];
  __shared__ unsigned char s_f[T_];
  __shared__ float r1[256], r2[256];
  __shared__ float s_b[2];
  __shared__ int   s_fp;

  const int b = blockIdx.x, tid = threadIdx.x;   /* blockDim = 256 */
  const float* P = ws + WS_P + b * T_;
  const float* Q = ws + WS_Q + b * T_;
  const int*   FLG = (const int*)(ws + WS_FLG) + b * T_;
  const float  cc = ws[WS_CC];

  /* ---- first_pad and N = sum(mask)-1 ---- */
  float lp = 0.f;
  for (int t = tid; t < T_; t += 256) { lp += (float)FLG[t]; s_p[t] = P[t]; }
  r1[tid] = lp;
  float lm = 0.f;
  for (int i = tid; i < 200; i += 256) lm += mask[b * 200 + i];
  r2[tid] = lm;
  __syncthreads();
  for (int off = 128; off; off >>= 1) {
    if (tid < off) { r1[tid] += r1[tid + off]; r2[tid] += r2[tid + off]; }
    __syncthreads();
  }
  if (tid == 0) { s_fp = (int)r1[0]; s_b[0] = r2[0] - 1.f; }
  __syncthreads();
  const int first_pad = s_fp;
  const float Nval = s_b[0];

  /* ---- raw weights: w = clip(relu(q - windowmean(p) + c), 0, 1) ---- */
  float lsum = 0.f;
  for (int t = tid; t < T_; t += 256) {
    float hv;
    if (t == 0) hv = s_p[0];                     /* hist row 0 := x[0] */
    else {
      int start = t - 10; if (start < 0) start = 0;
      float ssum = 0.f;
      for (int j = start; j < t; ++j) ssum += s_p[j];
      hv = ssum / (float)(t - start);
    }
    float w = Q[t] - hv + cc;
    w = fminf(fmaxf(w, 0.f), 1.0f);
    s_w[t] = w;
    out[O_ORI + b * T_ + t] = w;                 /* ori_weight */
    lsum += w;
  }
  __syncthreads();
  r1[tid] = lsum;
  __syncthreads();
  for (int off = 128; off; off >>= 1) {
    if (tid < off) r1[tid] += r1[tid + off];
    __syncthreads();
  }
  if (tid == 0) {
    float sa = r1[0];
    out[O_SUMA + b] = sa;                        /* sum_a */
    float sc = 1.f;
    if (is_training[0])
      sc = (sa == 0.f) ? 0.f : Nval / fmaxf(sa, 1e-8f);
    s_b[1] = sc;
  }
  __syncthreads();
  const float sc = s_b[1];
  for (int t = tid; t < T_; t += 256) {
    float w = sc * s_w[t];
    if (t > first_pad) w = 0.f;                  /* pad mask */
    s_w[t] = w;
  }
  __syncthreads();

  /* ---- scalar fire recurrence (serial, tiny) ---- */
  if (tid == 0) {
    float prev = 0.f;
    for (int t = 0; t < T_; ++t) {
      float w = s_w[t];
      int fired = (prev + w) > 1.0f;
      s_f[t] = (unsigned char)fired;
      prev = fired ? (w - (1.0f - prev)) : (w + prev);
    }
  }
  __syncthreads();
  for (int t = tid; t < T_; t += 256)
    out[O_FF + b * T_ + t] = s_f[t] ? 1.f : 0.f; /* fired_flag */

  /* ---- zero this batch's cif + not_padding slices ---- */
  {
    float* cifb = out + O_CIF + (size_t)b * L_ * H_;
    for (int i = tid; i < L_ * H_; i += 256) cifb[i] = 0.f;
    float* npb = out + O_NP + b * L_;
    for (int l = tid; l < L_; l += 256) npb[l] = 0.f;
  }
  __syncthreads();

  /* ---- vector integrate (serial in t, parallel in h) ---- */
  const int h = tid;
  const float* xb = x + (size_t)b * T_ * H_;
  float prev_s = xb[h];                          /* init carry = x[:,0,:] */
  float xv = xb[h];
  int nm = 0;                                    /* uniform across block */
  for (int t = 0; t < T_; ++t) {
    float xn = (t + 1 < T_) ? xb[(size_t)(t + 1) * H_ + h] : 0.f;  /* prefetch */
    const float w = s_w[t];
    const float integ = prev_s + (1.f - w) * xv;
    const int fired = s_f[t];
    if (fired && t <= first_pad) {               /* uniform branch */
      r1[tid] = fabsf(integ); r2[tid] = integ * integ;
      __syncthreads();
      for (int off = 128; off; off >>= 1) {
        if (tid < off) { r1[tid] += r1[tid + off]; r2[tid] += r2[tid + off]; }
        __syncthreads();
      }
      const float asum = r1[0], ssq = r2[0];
      __syncthreads();
      if (asum != 0.f) {
        const float inv = 1.f / fmaxf(sqrtf(ssq), 1e-12f);
        out[O_CIF + ((size_t)b * L_ + nm) * H_ + h] = integ * inv;
        if (tid == 0) out[O_NP + b * L_ + nm] = 1.f;
        nm++;
      }
    }
    prev_s = fired ? xv : integ;
    xv = xn;
  }
  /* ---- appended final state row ---- */
  {
    const float integ = prev_s;
    r1[tid] = fabsf(integ); r2[tid] = integ * integ;
    __syncthreads();
    for (int off = 128; off; off >>= 1) {
      if (tid < off) { r1[tid] += r1[tid + off]; r2[tid] += r2[tid + off]; }
      __syncthreads();
    }
    const float asum = r1[0], ssq = r2[0];
    __syncthreads();
    if (asum != 0.f) {
      const float inv = 1.f / fmaxf(sqrtf(ssq), 1e-12f);
      out[O_CIF + ((size_t)b * L_ + nm) * H_ + h] = integ * inv;
      if (tid == 0) out[O_NP + b * L_ + nm] = 1.f;
    }
  }
}

/* ------------------------------------------------------------------ */
extern "C" void kernel_launch(void* const* d_in, const int* in_sizes, int n_in,
                              void* d_out, int out_size, void* d_ws, size_t ws_size,
                              hipStream_t stream) {
  const float* x    = (const float*)d_in[0];   /* (32,2048,256) */
  const float* mask = (const float*)d_in[1];   /* (32,200,1)    */
  const float* W1   = (const float*)d_in[2];   /* (512,512)     */
  const float* b1   = (const float*)d_in[3];   /* (512,)        */
  const float* W2   = (const float*)d_in[4];   /* (1,512)       */
  const float* b2   = (const float*)d_in[5];   /* (1,)          */
  const int* is_training = (const int*)d_in[6];

  float* ws  = (float*)d_ws;
  float* out = (float*)d_out;

  cif_prep<<<1, 512, 0, stream>>>(W1, b1, W2, b2, ws);
  cif_dots<<<(B_ * T_) / 16, 32, 0, stream>>>(x, ws);
  cif_scan<<<B_, 256, 0, stream>>>(x, mask, is_training, ws, out);
}